// SimpleShader_13039520710672
// MI455X (gfx1250) — compile-verified
//
#include <hip/hip_runtime.h>
#include <cstdint>

// ---------------- constants from the reference ----------------
#define SIGMA_F 1e-4f
#define GAMMA_F 1e-4f
#define ZNEAR_F 1.0f
#define ZFAR_F  100.0f
#define EPS_F   1e-10f

constexpr int KF  = 8;    // faces per pixel
constexpr int TPB = 256;  // pixels (threads) per block

// ---------------- TDM descriptor plumbing ----------------
typedef unsigned int u32x4 __attribute__((ext_vector_type(4)));
typedef int          i32x4 __attribute__((ext_vector_type(4)));
typedef int          i32x8 __attribute__((ext_vector_type(8)));

// 1-D contiguous DMA: nelem 32-bit words from global -> LDS.
// D# layout per CDNA5 ISA ch.8 (group0 128b, group1 256b), little-endian dwords.
__device__ __forceinline__ void tdm_copy_1d(uint64_t gaddr, uint32_t lds_addr,
                                            uint32_t nelem)
{
    u32x4 g0;
    g0[0] = 1u;                                   // count=1 (valid user D#)
    g0[1] = lds_addr;                             // lds_addr [63:32]
    g0[2] = (uint32_t)(gaddr & 0xFFFFFFFFull);    // global_addr[31:0]
    g0[3] = ((uint32_t)(gaddr >> 32) & 0x01FFFFFFu) | (2u << 30); // [56:32]+type=2

    i32x8 g1;
    g1[0] = (int)(2u << 16);                      // workgroup_mask=0, data_size=4B
    g1[1] = (int)((nelem & 0xFFFFu) << 16);       // tensor_dim0[15:0] (hi half)
    g1[2] = (int)(((nelem >> 16) & 0xFFFFu)       // tensor_dim0[31:16]
                  | (1u << 16));                  // tensor_dim1 = 1
    g1[3] = (int)((nelem & 0xFFFFu) << 16);       // tile_dim0 = nelem (1-D tile)
    g1[4] = 0;                                    // tile_dim1=0, tile_dim2=0
    g1[5] = (int)nelem;                           // tensor_dim0_stride lo
    g1[6] = 0;                                    // stride hi / dim1_stride lo
    g1[7] = 0;

    i32x4 gz4 = {0, 0, 0, 0};                     // groups 2/3 unused (<=2D)
    i32x8 gz8 = {0, 0, 0, 0, 0, 0, 0, 0};         // trailing group (unused)
    __builtin_amdgcn_tensor_load_to_lds(g0, g1, gz4, gz4, gz8, /*cpol*/0);
}

// ---------------- kernel ----------------
__global__ __launch_bounds__(TPB)
void soft_blend_kernel(const int*   __restrict__ p2f,
                       const float* __restrict__ bary,
                       const float* __restrict__ dists,
                       const float* __restrict__ zbuf,
                       const float* __restrict__ fcol,
                       float*       __restrict__ out)
{
    __shared__ int   s_p2f [TPB * KF];        //  8 KB
    __shared__ float s_dist[TPB * KF];        //  8 KB
    __shared__ float s_zbuf[TPB * KF];        //  8 KB
    __shared__ float s_bary[TPB * KF * 3];    // 24 KB

    const uint32_t blk = blockIdx.x;

    // Wave 0 only issues the DMA. TDM ignores EXEC, so force a *scalar*
    // branch (uniform SGPR condition) so waves 1..7 jump over the issue.
    const unsigned wave_id = __builtin_amdgcn_readfirstlane(threadIdx.x >> 5);
    if (wave_id == 0) {
        const size_t base  = (size_t)blk * TPB * KF;       // k-major element base
        tdm_copy_1d((uint64_t)(uintptr_t)(p2f   + base),
                    (uint32_t)(uintptr_t)s_p2f,  TPB * KF);
        tdm_copy_1d((uint64_t)(uintptr_t)(dists + base),
                    (uint32_t)(uintptr_t)s_dist, TPB * KF);
        tdm_copy_1d((uint64_t)(uintptr_t)(zbuf  + base),
                    (uint32_t)(uintptr_t)s_zbuf, TPB * KF);
        tdm_copy_1d((uint64_t)(uintptr_t)(bary  + base * 3),
                    (uint32_t)(uintptr_t)s_bary, TPB * KF * 3);
        __builtin_amdgcn_s_wait_tensorcnt(0);
    }
    __syncthreads();

    const int t = threadIdx.x;

    // ---- pass 1: coverage prob, normalized inverse depth, running max ----
    float prob[KF], zin[KF];
    int   face[KF];
    float one_minus = 1.0f;
    float zmax = EPS_F;
#pragma unroll
    for (int k = 0; k < KF; ++k) {
        const int   f = s_p2f [t * KF + k];
        const float d = s_dist[t * KF + k];
        const float z = s_zbuf[t * KF + k];
        const float m = (f >= 0) ? 1.0f : 0.0f;
        face[k] = f;
        // sigmoid(-d/sigma) * mask
        const float p = m / (1.0f + __expf(d * (1.0f / SIGMA_F)));
        prob[k]    = p;
        one_minus *= (1.0f - p);
        const float zi = (ZFAR_F - z) * (1.0f / (ZFAR_F - ZNEAR_F)) * m;
        zin[k] = zi;
        zmax   = fmaxf(zmax, zi);
    }
    const float alpha = 1.0f - one_minus;

    // ---- softmax-over-depth weights ----
    const float delta = __expf((EPS_F - zmax) * (1.0f / GAMMA_F));
    float wnum[KF];
    float denom = delta;
#pragma unroll
    for (int k = 0; k < KF; ++k) {
        const float w = prob[k] * __expf((zin[k] - zmax) * (1.0f / GAMMA_F));
        wnum[k] = w;
        denom  += w;
    }
    const float inv_den = 1.0f / denom;

    // ---- pass 2: gather face colors only for surviving weights ----
    float r = 0.0f, g = 0.0f, b = 0.0f;
#pragma unroll
    for (int k = 0; k < KF; ++k) {
        const float w = wnum[k] * inv_den;
        if (w > 0.0f) {                       // gamma=1e-4 => usually 1 of 8
            const float* c  = fcol + (size_t)face[k] * 9;  // L2-resident
            const float  b0 = s_bary[t * (KF * 3) + 3 * k + 0];
            const float  b1 = s_bary[t * (KF * 3) + 3 * k + 1];
            const float  b2 = s_bary[t * (KF * 3) + 3 * k + 2];
            r = fmaf(w, fmaf(b0, c[0], fmaf(b1, c[3], b2 * c[6])), r);
            g = fmaf(w, fmaf(b0, c[1], fmaf(b1, c[4], b2 * c[7])), g);
            b = fmaf(w, fmaf(b0, c[2], fmaf(b1, c[5], b2 * c[8])), b);
        }
    }
    // BG = (0,0,0) so the background term adds nothing to rgb.

    const size_t pix = (size_t)blk * TPB + t;
    reinterpret_cast<float4*>(out)[pix] = make_float4(r, g, b, alpha);
}

// ---------------- launch ----------------
extern "C" void kernel_launch(void* const* d_in, const int* in_sizes, int n_in,
                              void* d_out, int out_size, void* d_ws, size_t ws_size,
                              hipStream_t stream)
{
    const int*   p2f  = (const int*)  d_in[0];  // (N,H,W,K)
    const float* bary = (const float*)d_in[1];  // (N,H,W,K,3)
    const float* dist = (const float*)d_in[2];  // (N,H,W,K)
    const float* zb   = (const float*)d_in[3];  // (N,H,W,K)
    const float* fc   = (const float*)d_in[4];  // (F,3,3)
    float* out        = (float*)d_out;          // (N,H,W,4)

    const int npix = in_sizes[0] / KF;          // N*H*W (divisible by TPB)
    const int grid = npix / TPB;

    soft_blend_kernel<<<grid, TPB, 0, stream>>>(p2f, bary, dist, zb, fc, out);
}